// WanS2VRotaryPosEmbed_30605936951392
// MI455X (gfx1250) — compile-verified
//
#include <hip/hip_runtime.h>
#include <math.h>

// WanS2V rotary position embedding table for gfx1250 (MI455X).
//
// Reference semantics (shapes fixed by setup_inputs):
//   ppf=21, pph=24, ppw=40 ; image rows: 1 x 24 x 40 at temporal pos 30
//   S = 20160 + 960 = 21120 rows, each row = 64 angles:
//     k in [0,22):  ang = t_pos * theta^(-2k/44)
//     k in [22,43): ang = h     * theta^(-2(k-22)/42)
//     k in [43,64): ang = w     * theta^(-2(k-43)/42)
//   out[0, s, head, k, 0:2] = (cos ang, sin ang), broadcast over 40 heads.
//   Output: 1*21120*40*64*2 f32 = 432.5 MB -> pure store-bandwidth problem
//   (23.3 TB/s HBM => ~18.6 us floor). Input VALUES are unused.
//
// CDNA5 strategy: compute each row's 512B (cos,sin) pattern once into LDS,
// then replicate it 40x straight from LDS with the gfx1250 async LDS->memory
// engine: GLOBAL_STORE_ASYNC_FROM_LDS_B128 (ASYNCcnt-tracked, no VGPR data
// round-trip), with th:TH_STORE_NT since the write-only 432MB stream is
// 2.25x the 192MB L2.

typedef float v4f __attribute__((ext_vector_type(4)));
typedef float v2f __attribute__((ext_vector_type(2)));

#define NH     40          // N_HEADS
#define HD2    64          // HEAD_DIM / 2
#define PPF    21
#define PPH    24
#define PPW    40
#define SV     (PPF * PPH * PPW)   // 20160 video rows
#define SI     (1 * PPH * PPW)     // 960 image rows (t pos = 30)
#define STOT   (SV + SI)           // 21120
#define IMG_T  30

#define BS       4         // s-rows per block
#define THREADS  256       // 8 wave32 waves
#define ROW_BYTES (NH * HD2 * 2 * 4)   // 20480 B per s-row

#define LOG2_THETA 13.287712379549449f   // log2(10000)

__global__ __launch_bounds__(THREADS)
void wan_s2v_rope_kernel(float* __restrict__ out)
{
    // Per-block staging: BS rows x 64 (cos,sin) float2 pairs = BS x 512 B.
    __shared__ v4f tile[BS * (HD2 / 2)];   // 2 KB

    const int tid = threadIdx.x;

    // ---- Phase 1: compute 4 rows x 64 angles; one (s_local, k) per thread ----
    {
        const int s_local = tid >> 6;         // 0..3
        const int k       = tid & 63;         // 0..63
        const int s       = blockIdx.x * BS + s_local;

        if (s < STOT) {
            int f, h, w;
            if (s < SV) {
                f = s / (PPH * PPW);
                const int r = s - f * (PPH * PPW);
                h = r / PPW;
                w = r - h * PPW;
            } else {
                const int r = s - SV;
                f = IMG_T;
                h = r / PPW;
                w = r - h * PPW;
            }

            float pos, ex;
            if (k < 22) {                     // temporal block, dim = 44
                pos = (float)f;
                ex  = -(2.0f * (float)k) * (1.0f / 44.0f);
            } else if (k < 43) {              // height block, dim = 42
                pos = (float)h;
                ex  = -(2.0f * (float)(k - 22)) * (1.0f / 42.0f);
            } else {                          // width block, dim = 42
                pos = (float)w;
                ex  = -(2.0f * (float)(k - 43)) * (1.0f / 42.0f);
            }

            const float inv = exp2f(ex * LOG2_THETA);   // theta^(-2j/dim) -> v_exp_f32
            const float ang = pos * inv;

            float sn, cs;
            __sincosf(ang, &sn, &cs);                   // v_sin_f32 / v_cos_f32

            v2f p; p.x = cs; p.y = sn;
            reinterpret_cast<v2f*>(tile)[s_local * HD2 + k] = p;
        }
    }

    __syncthreads();

    // ---- Phase 2: async LDS->memory broadcast over 40 heads -----------------
    // Per row: 1280 x 16B = 20 KB; 256 lanes x 5 async b128 stores. Each lane:
    // 64-bit global address (one base + offset: immediates) + 32-bit LDS addr.
    // ISA 10.2: low 32 bits of a generic LDS pointer == wave-relative LDS addr.
    const unsigned lane_pat = (unsigned)(tid & 31);     // which 16B of pattern
    const unsigned lds_base = (unsigned)(size_t)(&tile[0]);

    #pragma unroll
    for (int sl = 0; sl < BS; ++sl) {
        const unsigned l = lds_base + (sl * 32 + lane_pat) * 16u;
        const unsigned long long g =
            (unsigned long long)(size_t)out
            + (unsigned long long)(blockIdx.x * BS + sl) * (unsigned long long)ROW_BYTES
            + (unsigned long long)(tid * 16);

        asm volatile("global_store_async_from_lds_b128 %0, %1, off th:TH_STORE_NT"
                     :: "v"(g), "v"(l) : "memory");
        asm volatile("global_store_async_from_lds_b128 %0, %1, off offset:4096 th:TH_STORE_NT"
                     :: "v"(g), "v"(l) : "memory");
        asm volatile("global_store_async_from_lds_b128 %0, %1, off offset:8192 th:TH_STORE_NT"
                     :: "v"(g), "v"(l) : "memory");
        asm volatile("global_store_async_from_lds_b128 %0, %1, off offset:12288 th:TH_STORE_NT"
                     :: "v"(g), "v"(l) : "memory");
        asm volatile("global_store_async_from_lds_b128 %0, %1, off offset:16384 th:TH_STORE_NT"
                     :: "v"(g), "v"(l) : "memory");
    }

    // Drain the async engine before LDS can be reallocated (S_ENDPGM also does
    // an implicit wait-idle; this makes the dependency explicit).
    asm volatile("s_wait_asynccnt 0x0" ::: "memory");
}

extern "C" void kernel_launch(void* const* d_in, const int* in_sizes, int n_in,
                              void* d_out, int out_size, void* d_ws, size_t ws_size,
                              hipStream_t stream)
{
    (void)d_in; (void)in_sizes; (void)n_in;   // reference uses only input SHAPES
    (void)d_ws; (void)ws_size; (void)out_size;

    float* out = reinterpret_cast<float*>(d_out);

    const int grid = (STOT + BS - 1) / BS;    // 5280 blocks, exact
    wan_s2v_rope_kernel<<<grid, THREADS, 0, stream>>>(out);
}